// DebedderNeuron_45981919871511
// MI455X (gfx1250) — compile-verified
//
#include <hip/hip_runtime.h>

typedef __attribute__((ext_vector_type(2))) float v2f;
typedef __attribute__((ext_vector_type(8))) float v8f;

// ---- static architecture constants (match the reference) ----
static constexpr int I_DIM = 370816;
static constexpr int X_CH  = 448;   // sum(KERNEL_NO)
static constexpr int D     = 512;   // D_MODEL (K of the GEMMs)

template<int LAYER> struct LCfg;
//                         K     CH    OFF   N      ISTART  W9C    BIASI    NEXT    NSTR
template<> struct LCfg<0>{ static constexpr int K=64,  CH=3,   OFF=0,   N=1180, ISTART=0,     W9C=27,   BIASI=1728,   NEXT=1792,  NSTR=576;  };
template<> struct LCfg<1>{ static constexpr int K=128, CH=64,  OFF=64,  N=2881, ISTART=1792,  W9C=576,  BIASI=75520,  NEXT=75648, NSTR=1152; };
template<> struct LCfg<2>{ static constexpr int K=256, CH=128, OFF=192, N=1153, ISTART=75648, W9C=1152, BIASI=370560, NEXT=0,     NSTR=1;    };

// One fused GEMM + analytic-scatter kernel per CNN layer.
// Wave tile: 32(M) x 64(N) via 2x4 V_WMMA_F32_16X16X4_F32 fragments.
// Block: 128 threads = 4 waves stacked along M -> WG tile 128x64.
template<int LAYER>
__global__ __launch_bounds__(128) void gemm_scatter(
    const float* __restrict__ x,     // [128, 448, 512]
    const float* __restrict__ W,     // [512, N]
    const float* __restrict__ bias,  // [N]
    float*       __restrict__ y)     // [128, I_DIM]
{
  using C = LCfg<LAYER>;
  const int lane  = threadIdx.x & 31;
  const int wave  = threadIdx.x >> 5;
  const int laneN = lane & 15;
  const int hi    = lane >> 4;               // which half-wave (K pair / M+8)

  const int n0 = blockIdx.x * 64;            // first output column of this WG
  const int r0 = (blockIdx.y * 4 + wave) * 32; // wave's first GEMM row (m = b*K + kd)

  // ---- per-lane A base pointers (A frag: M=laneN, K = hi*2 + {0,1} -> contiguous pair)
  const float* aptr[2];
#pragma unroll
  for (int mt = 0; mt < 2; ++mt) {
    const int gr = r0 + mt * 16 + laneN;     // global GEMM row
    const int b  = gr / C::K;                // K is a power of two -> shift
    const int kd = gr % C::K;
    aptr[mt] = x + (size_t)(b * X_CH + C::OFF + kd) * D + hi * 2;
  }

  // ---- per-lane B base pointers (B frag: N=laneN, K = hi*2 + {0,1} -> stride-N pair)
  const float* bptr[4];
#pragma unroll
  for (int nt = 0; nt < 4; ++nt) {
    int j = n0 + nt * 16 + laneN;
    if (j >= C::N) j = C::N - 1;             // clamp ragged edge (stores are guarded)
    bptr[nt] = W + (size_t)(hi * 2) * C::N + j;
  }

  v8f acc[2][4];
#pragma unroll
  for (int mt = 0; mt < 2; ++mt)
#pragma unroll
    for (int nt = 0; nt < 4; ++nt)
      acc[mt][nt] = {};

  // ---- K loop: one 16x16x4 WMMA step per iteration, 2x4 tiles
#pragma unroll 2
  for (int k0 = 0; k0 < D; k0 += 4) {
    v2f a[2], bb[4];
#pragma unroll
    for (int mt = 0; mt < 2; ++mt)
      a[mt] = *(const v2f*)(aptr[mt] + k0);  // 8B contiguous A pair
#pragma unroll
    for (int nt = 0; nt < 4; ++nt) {
      bb[nt].x = bptr[nt][k0 * C::N];
      bb[nt].y = bptr[nt][k0 * C::N + C::N];
    }
#pragma unroll
    for (int mt = 0; mt < 2; ++mt)
#pragma unroll
      for (int nt = 0; nt < 4; ++nt)
        acc[mt][nt] = __builtin_amdgcn_wmma_f32_16x16x4_f32(
            false, a[mt], false, bb[nt], (short)0, acc[mt][nt], false, false);
  }

  // ---- epilogue: bias add + analytic scatter into y
  // C/D layout: VGPR r -> M = r + 8*hi (lanes 16-31), N = laneN.
#pragma unroll
  for (int nt = 0; nt < 4; ++nt) {
    const int j = n0 + nt * 16 + laneN;
    if (j >= C::N) continue;
    const float bj = bias[j];
#pragma unroll
    for (int mt = 0; mt < 2; ++mt) {
#pragma unroll
      for (int r = 0; r < 8; ++r) {
        const int gr = r0 + mt * 16 + hi * 8 + r;
        const int b  = gr / C::K;
        const int kd = gr % C::K;
        const float v = acc[mt][nt][r] + bj;
        float* yrow = y + (size_t)b * I_DIM;
        if (j < C::W9C) {                    // layer's own kernel weights
          const int i = C::ISTART + kd * C::W9C + j;
          if      (LAYER == 0) yrow[i] = v;                      // written once
          else if (LAYER == 1) yrow[i] = (yrow[i] + v) * 0.5f;   // + layer0 ext, halved
          else                 yrow[i] = yrow[i] + v;            // + layer1 ext, no halve
        } else if (j == C::W9C) {            // layer's bias slot
          const int i = C::BIASI + kd;
          if (LAYER == 1) yrow[i] = v * 0.5f;  // reference halves whole middle slice
          else            yrow[i] = v;
        } else {                             // extension into next layer's weights
          const int e  = j - (C::W9C + 1);
          const int kn = e / 9;
          const int t  = e - kn * 9;
          const int i  = C::NEXT + kd * 9 + kn * C::NSTR + t;
          yrow[i] = v;                       // raw; next layer's kernel accumulates
        }
      }
    }
  }
}

extern "C" void kernel_launch(void* const* d_in, const int* in_sizes, int n_in,
                              void* d_out, int out_size, void* d_ws, size_t ws_size,
                              hipStream_t stream) {
  (void)in_sizes; (void)n_in; (void)out_size; (void)d_ws; (void)ws_size;
  const float* x  = (const float*)d_in[0];
  const float* W0 = (const float*)d_in[1];
  const float* b0 = (const float*)d_in[2];
  const float* W1 = (const float*)d_in[3];
  const float* b1 = (const float*)d_in[4];
  const float* W2 = (const float*)d_in[5];
  const float* b2 = (const float*)d_in[6];
  float* y = (float*)d_out;

  const dim3 blk(128);
  // grid.y = M/128 = (128*K)/128 = K ; grid.x = ceil(N/64)
  gemm_scatter<0><<<dim3((1180 + 63) / 64,  64), blk, 0, stream>>>(x, W0, b0, y);
  gemm_scatter<1><<<dim3((2881 + 63) / 64, 128), blk, 0, stream>>>(x, W1, b1, y);
  gemm_scatter<2><<<dim3((1153 + 63) / 64, 256), blk, 0, stream>>>(x, W2, b2, y);
}